// RelMHAtt_360777252951
// MI455X (gfx1250) — compile-verified
//
#include <hip/hip_runtime.h>

// ---------------------------------------------------------------------------
// Relative multi-head attention, fp32, CDNA5 (gfx1250, wave32).
//   B=4, N=512, H=16, D=64, HS=1024, RS=64
// Pipeline:
//   1) gemm_kernel x3 : q/k/v @ W  -> [B,H,N,D]   (V_WMMA_F32_16X16X4_F32,
//                       tiles staged with TENSOR_LOAD_TO_LDS / TENSORcnt)
//   2) attn_kernel    : fused rel-bias GEMM + flash softmax + P@V
//   3) gemm_kernel    : atted @ Wm -> out
// ---------------------------------------------------------------------------

typedef float v2f __attribute__((ext_vector_type(2)));
typedef float v8f __attribute__((ext_vector_type(8)));
typedef unsigned int u32;
typedef u32 v4u __attribute__((ext_vector_type(4)));
typedef int  v8i __attribute__((ext_vector_type(8)));
typedef int  v4i __attribute__((ext_vector_type(4)));

#define B_    4
#define N_    512
#define H_    16
#define D_    64
#define HS_   1024
#define SCALE_ 0.125f  // 1/sqrt(64)

// D = A(16x4) * B(4x16) + C, fp32.  A/B = 2 VGPRs, C/D = 8 VGPRs (wave32).
__device__ __forceinline__ v8f wmma4(v2f a, v2f b, v8f c) {
  return __builtin_amdgcn_wmma_f32_16x16x4_f32(
      /*neg_a=*/false, a, /*neg_b=*/false, b,
      /*c_mod=*/(short)0, c, /*reuse_a=*/false, /*reuse_b=*/false);
}

#if __has_builtin(__builtin_amdgcn_tensor_load_to_lds)
#define USE_TDM 1
// TDM 2D tile load: tile_h rows x tile_w f32 elements, row stride
// `stride_elems`, into LDS at byte offset `lds_byte`, inserting
// (pad_amt_code+1) dwords of LDS padding every 2^(pad_int_code+1) dwords
// (produces the bank-conflict-free pitches used by the compute loop).
// D# per CDNA5 ISA; tiles are always fully
// in-bounds so tensor_dim0/1 only need to exceed the tile extents.
// This toolchain's builtin is the 6-arg clang-23 form:
//   (v4u g0, v8i g1, v4i g2, v4i g3, v8i, i32 cpol)
__device__ __forceinline__ void tdm_load_2d(u32 lds_byte, const float* gp,
                                            u32 tile_w, u32 tile_h,
                                            u32 stride_elems,
                                            u32 pad_int_code,
                                            u32 pad_amt_code) {
  const unsigned long long ga = (unsigned long long)(uintptr_t)gp;
  v4u g0;
  g0[0] = 1u;                                  // count=1, user descriptor
  g0[1] = lds_byte;                            // lds_addr (bytes)
  g0[2] = (u32)(ga & 0xffffffffu);             // global_addr[31:0]
  g0[3] = (u32)((ga >> 32) & 0x01ffffffu) | (2u << 30);  // addr[56:32],type=2
  const u32 td0 = stride_elems;                // tensor_dim0 >= tile_w
  const u32 td1 = 4096u;                       // tensor_dim1 >= tile_h
  v8i g1;
  g1[0] = (int)((2u << 16) |                   // data_size = 4 bytes
                (1u << 20) |                   // pad_enable
                (pad_int_code << 22) | (pad_amt_code << 25));
  g1[1] = (int)((td0 & 0xffffu) << 16);        // tensor_dim0[15:0]
  g1[2] = (int)(((td0 >> 16) & 0xffffu) | ((td1 & 0xffffu) << 16));
  g1[3] = (int)(((td1 >> 16) & 0xffffu) | (tile_w << 16));  // tile_dim0
  g1[4] = (int)(tile_h & 0xffffu);             // tile_dim1 (tile_dim2 = 0)
  g1[5] = (int)stride_elems;                   // tensor_dim0_stride[31:0]
  g1[6] = 0;                                   // stride hi, dim1_stride lo
  g1[7] = 0;
  const v4i zz4 = {0, 0, 0, 0};
  const v8i zz8 = {0, 0, 0, 0, 0, 0, 0, 0};
  __builtin_amdgcn_tensor_load_to_lds(g0, g1, zz4, zz4, zz8, 0);
}
#endif

// ---------------------------------------------------------------------------
// GEMM: C[2048x1024] = A[2048x1024] @ Bw[1024x1024]
// Block: 128 threads (4 waves), tile 64(M) x 64(N), K-chunk 32.
// headsplit=1: write C in [B,H,N,D] layout (row -> (b,n), col -> (h,d)).
// LDS pitches 36/72: fragment reads are bank-conflict-free
// (A reads stride rows: bank=(4*row+col)%64 distinct; B pitch 72 ->
//  half0/half1 lane banks offset by 16, no overlap in a 16-lane group).
// ---------------------------------------------------------------------------
__global__ __launch_bounds__(128) void gemm_kernel(
    const float* __restrict__ A, const float* __restrict__ Bw,
    float* __restrict__ C, int headsplit) {
  __shared__ __align__(16) float As[64 * 36];
  __shared__ __align__(16) float Bs[32 * 72];
  const int t    = threadIdx.x;
  const int wave = t >> 5;
  const int lane = t & 31;
  const int half = lane >> 4;
  const int l16  = lane & 15;
  const int bm   = (blockIdx.x >> 4) * 64;  // 32 M tiles
  const int bn   = (blockIdx.x & 15) * 64;  // 16 N tiles

  v8f acc[4] = {};

  for (int kc = 0; kc < 1024; kc += 32) {
#ifdef USE_TDM
    // Tensor Data Mover stages both tiles; wave 0 issues (TDM ignores EXEC,
    // the branch keeps the other 3 waves from duplicating the DMA), then
    // waits TENSORcnt==0 before the whole group passes the barrier.
    if (t == 0) {
      // A: 64 rows x 32 elems, pad 4 dwords every 32  -> pitch 36
      tdm_load_2d((u32)(uintptr_t)&As[0], &A[(size_t)bm * 1024 + kc],
                  /*w=*/32, /*h=*/64, /*stride=*/1024, /*int=*/4, /*amt=*/3);
      // B: 32 rows x 64 elems, pad 8 dwords every 64  -> pitch 72
      tdm_load_2d((u32)(uintptr_t)&Bs[0], &Bw[(size_t)kc * 1024 + bn],
                  /*w=*/64, /*h=*/32, /*stride=*/1024, /*int=*/5, /*amt=*/7);
      __builtin_amdgcn_s_wait_tensorcnt(0);
    }
#else
    // Fallback: manual staging (512 float4 per tile, 4 per thread)
    #pragma unroll
    for (int i = 0; i < 4; ++i) {
      const int idx = t + i * 128;
      const int ar = idx >> 3, ac = (idx & 7) << 2;
      *(float4*)&As[ar * 36 + ac] =
          *(const float4*)&A[(size_t)(bm + ar) * 1024 + kc + ac];
      const int brow = idx >> 4, bc = (idx & 15) << 2;
      *(float4*)&Bs[brow * 72 + bc] =
          *(const float4*)&Bw[(size_t)(kc + brow) * 1024 + bn + bc];
    }
#endif
    __syncthreads();
    #pragma unroll
    for (int kk = 0; kk < 32; kk += 4) {
      v2f a;
      a.x = As[(wave * 16 + l16) * 36 + kk + 2 * half];
      a.y = As[(wave * 16 + l16) * 36 + kk + 2 * half + 1];
      #pragma unroll
      for (int j = 0; j < 4; ++j) {
        v2f b;
        b.x = Bs[(kk + 2 * half) * 72 + j * 16 + l16];
        b.y = Bs[(kk + 2 * half + 1) * 72 + j * 16 + l16];
        acc[j] = wmma4(a, b, acc[j]);
      }
    }
    __syncthreads();
  }

  // C-fragment: VGPR r, lanes 0-15 -> M=r, lanes 16-31 -> M=r+8, N=lane%16
  #pragma unroll
  for (int j = 0; j < 4; ++j) {
    #pragma unroll
    for (int r = 0; r < 8; ++r) {
      const int gm = bm + wave * 16 + r + 8 * half;
      const int gn = bn + j * 16 + l16;
      const float val = acc[j][r];
      if (headsplit) {
        const int b = gm >> 9, n = gm & 511, h = gn >> 6, d = gn & 63;
        C[(size_t)(((b * H_ + h) * N_) + n) * D_ + d] = val;
      } else {
        C[(size_t)gm * 1024 + gn] = val;
      }
    }
  }
}

// ---------------------------------------------------------------------------
// Fused attention. Grid: B * N/16 = 128 blocks, 512 threads = 16 waves.
// Wave w == head w. Waves cooperate on the rel-bias GEMM so rel_embed
// (268 MB, the bandwidth driver) is streamed from HBM exactly once:
//   wave w computes bias rows for qloc=w: rel(16kp x 64) @ Wr(64 x 16heads),
//   exchanged through LDS (pitch 17 -> conflict-free column reads).
// Then per-head flash attention over k-chunks of 16.
// ---------------------------------------------------------------------------
__global__ __launch_bounds__(512) void attn_kernel(
    const float* __restrict__ qh, const float* __restrict__ kh,
    const float* __restrict__ vh, const unsigned char* __restrict__ mask,
    const float* __restrict__ rel, const float* __restrict__ Wr,
    const float* __restrict__ brv, float* __restrict__ atted) {
  __shared__ float bias_s[16 * 16 * 17];  // [(qloc*16+kp)] x [head], pitch 17
  __shared__ float ps[16 * 16 * 17];      // [head][qloc] x [kp],     pitch 17

  const int t    = threadIdx.x;
  const int h    = t >> 5;           // wave id == head
  const int lane = t & 31;
  const int half = lane >> 4;
  const int l16  = lane & 15;
  const int b    = blockIdx.x >> 5;
  const int qt   = blockIdx.x & 31;  // 16-row q tile

  const float brl = brv[l16];        // bias for head column l16

  // Wr fragments (B matrix of bias GEMM), kept in VGPRs for the whole kernel
  v2f wrf[16];
  #pragma unroll
  for (int kk = 0; kk < 16; ++kk) {
    wrf[kk].x = Wr[(kk * 4 + 2 * half) * 16 + l16];
    wrf[kk].y = Wr[(kk * 4 + 2 * half + 1) * 16 + l16];
  }
  // Q fragments for this head's 16x64 tile
  const float* qbase = qh + (size_t)((b * H_ + h) * N_ + qt * 16 + l16) * 64;
  v2f qf[16];
  #pragma unroll
  for (int kk = 0; kk < 16; ++kk) {
    qf[kk].x = qbase[kk * 4 + 2 * half];
    qf[kk].y = qbase[kk * 4 + 2 * half + 1];
  }

  v8f acc[4] = {};
  float mi[8], li[8];
  #pragma unroll
  for (int r = 0; r < 8; ++r) { mi[r] = -3.0e38f; li[r] = 0.0f; }

  const float* khb = kh + (size_t)(b * H_ + h) * N_ * 64;
  const float* vhb = vh + (size_t)(b * H_ + h) * N_ * 64;
  // wave w owns rel rows for q = qt*16 + w
  const float* relb = rel + (size_t)((b * N_ + qt * 16 + h)) * N_ * 64;
  const unsigned char* mb = mask + (size_t)(b * N_ + qt * 16) * N_;

  for (int kc = 0; kc < N_; kc += 16) {
    // Prefetch next chunk of the rel stream (global_prefetch_b8); each lane
    // covers half of one 256B row of the upcoming 16x64 f32 tile.
    if (kc + 16 < N_) {
      __builtin_prefetch(relb + (size_t)(kc + 16 + l16) * 64 + half * 32, 0, 1);
    }

    // ---- bias tile: C[m=kp][n=head] = rel[q=h tile] @ Wr -------------------
    v8f cb = {};
    {
      const float* ra = relb + (size_t)(kc + l16) * 64;  // A: lane l16 = row kp
      #pragma unroll
      for (int kk = 0; kk < 16; ++kk) {
        v2f a;
        a.x = ra[kk * 4 + 2 * half];
        a.y = ra[kk * 4 + 2 * half + 1];
        cb = wmma4(a, wrf[kk], cb);
      }
    }
    #pragma unroll
    for (int r = 0; r < 8; ++r) {
      // relu -> clip(1e-6) -> log  ==  log(max(x + br, 1e-6))
      const float tb = fmaxf(cb[r] + brl, 1e-6f);
      bias_s[(h * 16 + (r + 8 * half)) * 17 + l16] = __logf(tb);
    }
    __syncthreads();

    // ---- scores S = Q @ K^T for my head -----------------------------------
    v8f s = {};
    #pragma unroll
    for (int kk = 0; kk < 16; ++kk) {
      v2f bf;  // B[d][kp] = kh[kp][d]
      bf.x = khb[(size_t)(kc + l16) * 64 + kk * 4 + 2 * half];
      bf.y = khb[(size_t)(kc + l16) * 64 + kk * 4 + 2 * half + 1];
      s = wmma4(qf[kk], bf, s);
    }
    #pragma unroll
    for (int r = 0; r < 8; ++r) {
      const int qloc = r + 8 * half;
      float sv = s[r] * SCALE_ + bias_s[(qloc * 16 + l16) * 17 + h];
      if (mb[(size_t)qloc * N_ + kc + l16]) sv = -1.0e9f;

      // online softmax, row reduction across the 16-lane half (cols kp)
      float rm = sv;
      rm = fmaxf(rm, __shfl_xor(rm, 1, 16));
      rm = fmaxf(rm, __shfl_xor(rm, 2, 16));
      rm = fmaxf(rm, __shfl_xor(rm, 4, 16));
      rm = fmaxf(rm, __shfl_xor(rm, 8, 16));
      const float Mn = fmaxf(mi[r], rm);
      const float sc = __expf(mi[r] - Mn);
      const float p  = __expf(sv - Mn);
      float rs = p;
      rs += __shfl_xor(rs, 1, 16);
      rs += __shfl_xor(rs, 2, 16);
      rs += __shfl_xor(rs, 4, 16);
      rs += __shfl_xor(rs, 8, 16);
      li[r] = li[r] * sc + rs;
      mi[r] = Mn;
      #pragma unroll
      for (int j = 0; j < 4; ++j) acc[j][r] *= sc;
      ps[(h * 16 + qloc) * 17 + l16] = p;  // stage P for fragment transpose
    }
    __syncthreads();

    // ---- O += P @ V (K = 16 this chunk) -----------------------------------
    #pragma unroll
    for (int kk = 0; kk < 4; ++kk) {
      v2f a;  // A[m=qloc][k=kp] from LDS
      a.x = ps[(h * 16 + l16) * 17 + kk * 4 + 2 * half];
      a.y = ps[(h * 16 + l16) * 17 + kk * 4 + 2 * half + 1];
      #pragma unroll
      for (int j = 0; j < 4; ++j) {
        v2f bf;  // B[kp][d], coalesced (lanes sweep d)
        bf.x = vhb[(size_t)(kc + kk * 4 + 2 * half) * 64 + j * 16 + l16];
        bf.y = vhb[(size_t)(kc + kk * 4 + 2 * half + 1) * 64 + j * 16 + l16];
        acc[j] = wmma4(a, bf, acc[j]);
      }
    }
  }

  // ---- normalize and write atted in [B,N,HS] layout for the final GEMM ----
  #pragma unroll
  for (int r = 0; r < 8; ++r) {
    const float inv = 1.0f / li[r];
    const int q = qt * 16 + r + 8 * half;
    #pragma unroll
    for (int j = 0; j < 4; ++j) {
      atted[(size_t)(b * N_ + q) * HS_ + h * 64 + j * 16 + l16] =
          acc[j][r] * inv;
    }
  }
}

// ---------------------------------------------------------------------------
extern "C" void kernel_launch(void* const* d_in, const int* in_sizes, int n_in,
                              void* d_out, int out_size, void* d_ws,
                              size_t ws_size, hipStream_t stream) {
  (void)in_sizes; (void)n_in; (void)out_size; (void)ws_size;
  const float* v  = (const float*)d_in[0];
  const float* k  = (const float*)d_in[1];
  const float* q  = (const float*)d_in[2];
  const unsigned char* mask = (const unsigned char*)d_in[3];  // bool, 1 byte
  const float* rel = (const float*)d_in[4];
  const float* Wv = (const float*)d_in[5];
  const float* Wk = (const float*)d_in[6];
  const float* Wq = (const float*)d_in[7];
  const float* Wr = (const float*)d_in[8];
  const float* br = (const float*)d_in[9];
  const float* Wm = (const float*)d_in[10];
  float* out = (float*)d_out;

  // workspace: qh | kh | vh | atted, each B*H*N*D = B*N*HS = 2,097,152 floats
  float* ws    = (float*)d_ws;
  float* qh    = ws;
  float* kh    = qh + 2097152;
  float* vh    = kh + 2097152;
  float* atted = vh + 2097152;

  const dim3 gg(512);  // (2048/64) * (1024/64)
  gemm_kernel<<<gg, 128, 0, stream>>>(q, Wq, qh, 1);
  gemm_kernel<<<gg, 128, 0, stream>>>(k, Wk, kh, 1);
  gemm_kernel<<<gg, 128, 0, stream>>>(v, Wv, vh, 1);
  attn_kernel<<<dim3(128), 512, 0, stream>>>(qh, kh, vh, mask, rel, Wr, br,
                                             atted);
  gemm_kernel<<<gg, 128, 0, stream>>>(atted, Wm, out, 0);
}